// MultiHeadAttentionLayer_42554535969392
// MI455X (gfx1250) — compile-verified
//
#include <hip/hip_runtime.h>

// Problem constants (match reference setup_inputs)
#define N_NODES 50000
#define N_EDGES 800000
#define IN_NODE 128
#define IN_EDGE 64
#define HEADS   8
#define OUT_DIM 8
#define HID     64   // HEADS * OUT_DIM

typedef float v2f __attribute__((ext_vector_type(2)));
typedef float v8f __attribute__((ext_vector_type(8)));

#define INV_SQRT8 0.35355339059327373f

// ---------------------------------------------------------------------------
// Zero the scatter accumulators (wV | z_sum | cnt are contiguous in ws)
// ---------------------------------------------------------------------------
__global__ void __launch_bounds__(256) zero_kernel(float* __restrict__ p, int n) {
    int i = blockIdx.x * 256 + threadIdx.x;
    if (i < n) p[i] = 0.0f;
}

// ---------------------------------------------------------------------------
// Node projections: Q/K/V = h @ W* + b*, fused (shared A loads).
// One wave per 16x16 output tile; K-loop over 128 in steps of 4 using
// V_WMMA_F32_16X16X4_F32.
//   A (16x4 f32): lane L holds row M=L&15; VGPR0/1 = K {0,1} (lanes 0-15)
//                 or K {2,3} (lanes 16-31).
//   B (4x16 f32): lane L holds col N=L&15; VGPR0/1 = K {0,1} / {2,3}.
//   D (16x16 f32): lane L holds col N=L&15; VGPR v = row v + 8*(L>>4).
// ---------------------------------------------------------------------------
__global__ void __launch_bounds__(256) node_proj_wmma(
    const float* __restrict__ h,
    const float* __restrict__ WQ, const float* __restrict__ bQ,
    const float* __restrict__ WK, const float* __restrict__ bK,
    const float* __restrict__ WV, const float* __restrict__ bV,
    float* __restrict__ Q, float* __restrict__ K, float* __restrict__ V)
{
    const int lane = threadIdx.x & 31;
    const int wave = threadIdx.x >> 5;
    const int tile = blockIdx.x * 8 + wave;
    const int numTiles = (N_NODES / 16) * (HID / 16);   // 12500
    if (tile >= numTiles) return;

    const int rowTile = tile >> 2;           // node tile
    const int colTile = tile & 3;            // 16-col tile of HID
    const int m   = lane & 15;
    const int kk  = (lane >> 4) << 1;        // K offset 0 or 2
    const int col = colTile * 16 + m;

    v8f cq = {}; v8f ck = {}; v8f cv = {};
    const float* __restrict__ arow = h + (size_t)(rowTile * 16 + m) * IN_NODE + kk;

    for (int k0 = 0; k0 < IN_NODE; k0 += 4) {
        v2f a;
        a.x = arow[k0];
        a.y = arow[k0 + 1];
        const int br = k0 + kk;
        v2f bq2; bq2.x = WQ[(size_t)br * HID + col]; bq2.y = WQ[(size_t)(br + 1) * HID + col];
        v2f bk2; bk2.x = WK[(size_t)br * HID + col]; bk2.y = WK[(size_t)(br + 1) * HID + col];
        v2f bv2; bv2.x = WV[(size_t)br * HID + col]; bv2.y = WV[(size_t)(br + 1) * HID + col];
        cq = __builtin_amdgcn_wmma_f32_16x16x4_f32(false, a, false, bq2, (short)0, cq, false, false);
        ck = __builtin_amdgcn_wmma_f32_16x16x4_f32(false, a, false, bk2, (short)0, ck, false, false);
        cv = __builtin_amdgcn_wmma_f32_16x16x4_f32(false, a, false, bv2, (short)0, cv, false, false);
    }

    const float biasq = bQ[col];
    const float biask = bK[col];
    const float biasv = bV[col];
    const int rbase = rowTile * 16 + ((lane >> 4) << 3);
#pragma unroll
    for (int v = 0; v < 8; ++v) {
        const size_t idx = (size_t)(rbase + v) * HID + col;
        Q[idx] = cq[v] + biasq;
        K[idx] = ck[v] + biask;
        V[idx] = cv[v] + biasv;
    }
}

// ---------------------------------------------------------------------------
// Fused edge pass. ONE WAVE handles a full 16-edge x 64-col tile:
//   - pe = e @ We + be via 4 independent WMMA accumulator chains (cols 0-15,
//     16-31, 32-47, 48-63) sharing the A (e-row) loads and edge_index loads.
//   - score = K[src]*Q[dst]/sqrt(8) + pe  -> written to e_out exactly once.
//   - per-(edge,head) a = exp(clip(sum_d score)) via aligned 8-lane butterfly
//     shuffles (head groups are lanes {0-7},{8-15},{16-23},{24-31}).
//   - atomic scatter of wV / z_sum / cnt by src. All register-array indices
//     are compile-time constants (no dynamic indexing -> no scratch).
// ---------------------------------------------------------------------------
__global__ void __launch_bounds__(256) edge_fused_wmma(
    const float* __restrict__ e,
    const float* __restrict__ We, const float* __restrict__ be,
    const long long* __restrict__ ei,          // [2, N_EDGES] int64
    const float* __restrict__ Q, const float* __restrict__ K,
    const float* __restrict__ V,
    float* __restrict__ e_out,
    float* __restrict__ wV, float* __restrict__ z_sum, float* __restrict__ cnt)
{
    const int lane = threadIdx.x & 31;
    const int wave = threadIdx.x >> 5;
    const int rowTile = blockIdx.x * 8 + wave;          // 16-edge tile
    if (rowTile >= N_EDGES / 16) return;                // 50000 tiles

    const int m   = lane & 15;
    const int kk  = (lane >> 4) << 1;                   // K offset 0 or 2

    // ---- WMMA GEMM: pe for all 4 column tiles, shared A loads ----
    v8f c[4] = {};
    const float* __restrict__ arow = e + (size_t)(rowTile * 16 + m) * IN_EDGE + kk;
    for (int k0 = 0; k0 < IN_EDGE; k0 += 4) {
        v2f a;
        a.x = arow[k0];
        a.y = arow[k0 + 1];
        const int br = k0 + kk;
        const float* __restrict__ w0 = We + (size_t)br * HID + m;
        const float* __restrict__ w1 = We + (size_t)(br + 1) * HID + m;
#pragma unroll
        for (int ct = 0; ct < 4; ++ct) {
            v2f b;
            b.x = w0[ct * 16];
            b.y = w1[ct * 16];
            c[ct] = __builtin_amdgcn_wmma_f32_16x16x4_f32(false, a, false, b,
                                                          (short)0, c[ct], false, false);
        }
    }

    // Per-lane biases for the 4 columns this lane owns (m, m+16, m+32, m+48)
    float bias[4];
#pragma unroll
    for (int ct = 0; ct < 4; ++ct) bias[ct] = be[ct * 16 + m];

    // ---- edge indices for this lane's 8 rows (shared across all 4 ct) ----
    const int rhalf = (lane >> 4) << 3;                 // 0 or 8
    long long srcs[8];
    long long dsts[8];
#pragma unroll
    for (int v = 0; v < 8; ++v) {
        const int edge = rowTile * 16 + rhalf + v;
        srcs[v] = ei[edge];
        dsts[v] = ei[N_EDGES + edge];
    }

    // ---- score = pe + K[src]*Q[dst]/sqrt(8); write e_out (in-place in c) ----
#pragma unroll
    for (int v = 0; v < 8; ++v) {
        const float* __restrict__ Kp = K + (size_t)srcs[v] * HID + m;
        const float* __restrict__ Qp = Q + (size_t)dsts[v] * HID + m;
        float* __restrict__ Ep = e_out + (size_t)(rowTile * 16 + rhalf + v) * HID + m;
#pragma unroll
        for (int ct = 0; ct < 4; ++ct) {
            const float score = c[ct][v] + bias[ct]
                              + Kp[ct * 16] * Qp[ct * 16] * INV_SQRT8;
            c[ct][v] = score;
            Ep[ct * 16] = score;
        }
    }

    // ---- per-(edge,head) sums: butterfly across each aligned 8-lane group;
    //      replace score with a = exp(clip(sum)) in place ----
#pragma unroll
    for (int ct = 0; ct < 4; ++ct) {
#pragma unroll
        for (int v = 0; v < 8; ++v) {
            float s = c[ct][v];
            s += __shfl_xor(s, 1, 32);
            s += __shfl_xor(s, 2, 32);
            s += __shfl_xor(s, 4, 32);
            s = fminf(fmaxf(s, -5.0f), 5.0f);
            c[ct][v] = __expf(s);
        }
    }

    // ---- scatter wV: each lane owns 4 columns x 8 rows ----
#pragma unroll
    for (int v = 0; v < 8; ++v) {
        const size_t srow = (size_t)srcs[v] * HID + m;
#pragma unroll
        for (int ct = 0; ct < 4; ++ct) {
            atomicAdd(&wV[srow + ct * 16], V[srow + ct * 16] * c[ct][v]);
        }
    }

    // ---- scatter z_sum: one designated lane per 8-lane head group does the
    //      8 rows with static indices (values are group-uniform after the
    //      butterfly). Group g covers head ct*2 + hg for rows rhalf..rhalf+7.
    if ((lane & 7) == 0) {
        const int hg = (lane >> 3) & 1;
#pragma unroll
        for (int ct = 0; ct < 4; ++ct) {
            const int head = ct * 2 + hg;
#pragma unroll
            for (int v = 0; v < 8; ++v) {
                atomicAdd(&z_sum[(size_t)srcs[v] * HEADS + head], c[ct][v]);
            }
        }
    }

    // ---- scatter cnt: exactly once per edge (lanes 0 and 16) ----
    if ((lane & 15) == 0) {
#pragma unroll
        for (int v = 0; v < 8; ++v) {
            atomicAdd(&cnt[srcs[v]], 1.0f);
        }
    }
}

// ---------------------------------------------------------------------------
// h_out = wV / (z_sum / max(cnt,1) + 1e-6)
// ---------------------------------------------------------------------------
__global__ void __launch_bounds__(256) finalize_kernel(
    const float* __restrict__ wV, const float* __restrict__ z_sum,
    const float* __restrict__ cnt, float* __restrict__ h_out)
{
    const int i = blockIdx.x * 256 + threadIdx.x;
    if (i >= N_NODES * HID) return;
    const int n = i >> 6;                // node
    const int head = (i >> 3) & 7;
    const float cmax = fmaxf(cnt[n], 1.0f);
    const float z = z_sum[n * HEADS + head] / cmax;
    h_out[i] = wV[i] / (z + 1e-6f);
}

// ---------------------------------------------------------------------------
extern "C" void kernel_launch(void* const* d_in, const int* in_sizes, int n_in,
                              void* d_out, int out_size, void* d_ws, size_t ws_size,
                              hipStream_t stream) {
    const long long* ei = (const long long*)d_in[0];
    const float* h  = (const float*)d_in[1];
    const float* e  = (const float*)d_in[2];
    const float* WQ = (const float*)d_in[3];
    const float* bQ = (const float*)d_in[4];
    const float* WK = (const float*)d_in[5];
    const float* bK = (const float*)d_in[6];
    const float* WV = (const float*)d_in[7];
    const float* bV = (const float*)d_in[8];
    const float* We = (const float*)d_in[9];
    const float* be = (const float*)d_in[10];

    float* out   = (float*)d_out;
    float* h_out = out;                               // [N_NODES * HID]
    float* e_out = out + (size_t)N_NODES * HID;       // [N_EDGES * HID]

    // Workspace layout (floats)
    float* ws   = (float*)d_ws;
    const size_t NH = (size_t)N_NODES * HID;          // 3.2M
    float* Q     = ws;                                // NH
    float* K     = Q + NH;                            // NH
    float* V     = K + NH;                            // NH
    float* wV    = V + NH;                            // NH
    float* z_sum = wV + NH;                           // N_NODES*HEADS
    float* cnt   = z_sum + (size_t)N_NODES * HEADS;   // N_NODES

    // 1) zero accumulators (wV | z_sum | cnt contiguous)
    const int nzero = (int)(NH + (size_t)N_NODES * HEADS + N_NODES);
    zero_kernel<<<(nzero + 255) / 256, 256, 0, stream>>>(wV, nzero);

    // 2) node projections Q/K/V (WMMA f32 16x16x4)
    {
        const int numTiles = (N_NODES / 16) * (HID / 16);   // 12500 tiles
        const int blocks = (numTiles + 7) / 8;              // 8 waves/block
        node_proj_wmma<<<blocks, 256, 0, stream>>>(h, WQ, bQ, WK, bK, WV, bV, Q, K, V);
    }

    // 3) fused edge pass: one wave per 16-edge tile, all 64 columns
    {
        const int numTiles = N_EDGES / 16;                  // 50000 tiles
        const int blocks = (numTiles + 7) / 8;
        edge_fused_wmma<<<blocks, 256, 0, stream>>>(e, We, be, ei, Q, K, V,
                                                    e_out, wV, z_sum, cnt);
    }

    // 4) finalize h_out
    {
        const int n = N_NODES * HID;
        finalize_kernel<<<(n + 255) / 256, 256, 0, stream>>>(wV, z_sum, cnt, h_out);
    }
}